// GCNAEAlpha_2207613190406
// MI455X (gfx1250) — compile-verified
//
#include <hip/hip_runtime.h>
#include <hip/hip_bf16.h>

// ---------------------------------------------------------------------------
// GCN-AE alpha graph conv + inner-product decoder for MI455X (gfx1250).
// Dominant cost: adj = rst @ rst^T  (16384^2 x 64 -> 1 GiB fp32 output).
// rst (4 MB) stays resident in the 192 MB L2; HBM traffic ~= the 1 GiB NT
// store stream, so we use fp32 WMMA (exact) + nontemporal stores.
// ---------------------------------------------------------------------------

#define F_IN 256
#define HID  64

typedef float v2f __attribute__((ext_vector_type(2)));
typedef float v8f __attribute__((ext_vector_type(8)));

// ---- degree accumulation ---------------------------------------------------
__global__ void deg_kernel(const int* __restrict__ src, const int* __restrict__ dst,
                           float* __restrict__ out_deg, float* __restrict__ in_deg, int E) {
    int e = blockIdx.x * blockDim.x + threadIdx.x;
    if (e >= E) return;
    atomicAdd(&out_deg[src[e]], 1.0f);
    atomicAdd(&in_deg[dst[e]],  1.0f);
}

// ---- h = (features * rsqrt(out_deg)) @ W  via V_WMMA_F32_16X16X4_F32 -------
// One wave32 computes one 16x16 tile of h.  Grid is exact: EXEC all ones.
__global__ __launch_bounds__(256) void proj_kernel(const float* __restrict__ feat,
                                                   const float* __restrict__ W,
                                                   const float* __restrict__ out_deg,
                                                   float* __restrict__ h) {
    const int wave   = threadIdx.x >> 5;
    const int lane   = threadIdx.x & 31;
    const int tileId = blockIdx.x * 8 + wave;
    const int tm = tileId >> 2;          // 1024 M-tiles
    const int tn = tileId & 3;           // 4 N-tiles (HID=64)
    const int l  = lane & 15;
    const int hi = (lane >= 16) ? 1 : 0; // half-wave select per ISA A/B layout

    const int row = tm * 16 + l;
    const float rs = rsqrtf(fmaxf(out_deg[row], 1.0f));
    const float* __restrict__ arow = feat + (size_t)row * F_IN;

    v8f c = {};
#pragma unroll 8
    for (int kk = 0; kk < F_IN; kk += 4) {
        const int k0 = kk + (hi ? 2 : 0);
        const int k1 = kk + (hi ? 3 : 1);
        v2f a, b;
        a.x = arow[k0] * rs;                        // A 16x4: lanes0-15 K0/K1, 16-31 K2/K3
        a.y = arow[k1] * rs;
        b.x = W[(size_t)k0 * HID + tn * 16 + l];    // B 4x16: row striped across lanes
        b.y = W[(size_t)k1 * HID + tn * 16 + l];
        c = __builtin_amdgcn_wmma_f32_16x16x4_f32(false, a, false, b,
                                                  (short)0, c, false, false);
    }
    float* __restrict__ hb = h + (size_t)(tm * 16) * HID + tn * 16;
#pragma unroll
    for (int r = 0; r < 8; ++r) {
        const int m = r + (hi ? 8 : 0);             // C layout: VGPR r -> M=r / M=8+r
        hb[(size_t)m * HID + l] = c[r];
    }
}

// ---- per-edge alpha select + scatter-add ----------------------------------
__global__ void edge_kernel(const float* __restrict__ h, const float* __restrict__ alpha,
                            const int* __restrict__ src, const int* __restrict__ dst,
                            const int* __restrict__ node_id, float* __restrict__ agg,
                            int E, int gene_num) {
    int t = blockIdx.x * blockDim.x + threadIdx.x;
    int e  = t >> 4;                 // 16 threads per edge
    int f0 = (t & 15) * 4;           // 4 features each
    if (e >= E) return;
    const int s = src[e], d = dst[e];
    const int sid = node_id[s], did = node_id[d];
    int idx = gene_num + 1;
    if (sid >= 0 && did < 0)        idx = sid;
    else if (did >= 0 && sid < 0)   idx = did;
    else if (did >= 0 && sid >= 0)  idx = gene_num;
    const float a = alpha[idx];
    const float* __restrict__ hs = h   + (size_t)s * HID + f0;
    float*       __restrict__ ad = agg + (size_t)d * HID + f0;
#pragma unroll
    for (int j = 0; j < 4; ++j)
        atomicAdd(ad + j, hs[j] * a);
}

// ---- rst = agg * rsqrt(in_deg) + bias -------------------------------------
__global__ void norm_kernel(const float* __restrict__ agg, const float* __restrict__ in_deg,
                            const float* __restrict__ bias, float* __restrict__ rst) {
    int t = blockIdx.x * blockDim.x + threadIdx.x;
    int n = t >> 6, j = t & 63;
    rst[t] = agg[t] * rsqrtf(fmaxf(in_deg[n], 1.0f)) + bias[j];
}

// ---- adj = rst @ rst^T  (128x64 LDS-tiled, fp32 WMMA, NT stores) ----------
// 256 threads = 8 wave32.  Each wave owns one N-tile (tn = wave&3), so its
// B fragments are loop-invariant: load them into registers ONCE, then sweep
// 4 M-tiles reading only A fragments from LDS.  LDS stride 68 floats keeps
// rows 16B-aligned and both access patterns bank-conflict-free.
#define LSTRIDE 68
__global__ __launch_bounds__(256) void gemm_kernel(const float* __restrict__ rst,
                                                   float* __restrict__ adj, int N) {
    __shared__ float As[128 * LSTRIDE];
    __shared__ float Bs[64 * LSTRIDE];
    const int bm = blockIdx.y;           // N/128 blocks
    const int bn = blockIdx.x;           // N/64 blocks
    const float* __restrict__ Ag = rst + (size_t)bm * 128 * HID;
    const float* __restrict__ Bg = rst + (size_t)bn * 64  * HID;

    // cooperative loads (rst is L2-resident: 4 MB << 192 MB)
#pragma unroll
    for (int i = 0; i < 8; ++i) {
        int lin = (threadIdx.x + i * 256) * 4;      // 128*64 floats
        int row = lin >> 6, k = lin & 63;
        *(float4*)&As[row * LSTRIDE + k] = *(const float4*)(Ag + lin);
    }
#pragma unroll
    for (int i = 0; i < 2; ++i) {
        int lin = (threadIdx.x + i * 256) * 4;      // 32*64 floats
        int row = lin >> 6, k = lin & 63;
        *(float4*)&Bs[row * LSTRIDE + k] = *(const float4*)(Bg + lin);
    }
    __syncthreads();

    const int wave = threadIdx.x >> 5;
    const int lane = threadIdx.x & 31;
    const int l  = lane & 15;
    const int hi = (lane >= 16) ? 1 : 0;
    const int tn = wave & 3;             // fixed N-tile per wave
    const int tb = wave >> 2;            // M-tile base offset (0 or 1)

    // hoist B fragments: B[k][n] = rstB[n][k]; lane l holds column n = tn*16+l
    v2f bfrag[16];
    {
        const float* __restrict__ bp = &Bs[(tn * 16 + l) * LSTRIDE];
#pragma unroll
        for (int s = 0; s < 16; ++s) {
            bfrag[s].x = bp[4 * s + (hi ? 2 : 0)];
            bfrag[s].y = bp[4 * s + (hi ? 3 : 1)];
        }
    }

#pragma unroll
    for (int i = 0; i < 4; ++i) {                   // 4 M-tiles per wave
        const int tm = 2 * i + tb;                  // covers 0..7 across waves
        const float* __restrict__ ap = &As[(tm * 16 + l) * LSTRIDE];
        v8f c = {};
#pragma unroll
        for (int s = 0; s < 16; ++s) {              // K = 64 in steps of 4
            v2f a;
            a.x = ap[4 * s + (hi ? 2 : 0)];         // A[m][k]
            a.y = ap[4 * s + (hi ? 3 : 1)];
            c = __builtin_amdgcn_wmma_f32_16x16x4_f32(false, a, false, bfrag[s],
                                                      (short)0, c, false, false);
        }
        float* __restrict__ ob = adj + (size_t)(bm * 128 + tm * 16) * N + bn * 64 + tn * 16;
#pragma unroll
        for (int r = 0; r < 8; ++r) {
            const int m = r + (hi ? 8 : 0);
            __builtin_nontemporal_store(c[r], ob + (size_t)m * N + l);  // 1 GiB stream: keep L2 for rst
        }
    }
}

extern "C" void kernel_launch(void* const* d_in, const int* in_sizes, int n_in,
                              void* d_out, int out_size, void* d_ws, size_t ws_size,
                              hipStream_t stream) {
    const float* features = (const float*)d_in[0];
    const float* W        = (const float*)d_in[1];
    const float* bias     = (const float*)d_in[2];
    const float* alpha    = (const float*)d_in[3];
    const int*   src      = (const int*)d_in[4];
    const int*   dst      = (const int*)d_in[5];
    const int*   node_id  = (const int*)d_in[6];

    const int hid      = in_sizes[2];            // 64
    const int fin      = in_sizes[1] / hid;      // 256
    const int N        = in_sizes[0] / fin;      // 16384
    const int E        = in_sizes[4];            // 524288
    const int gene_num = in_sizes[3] - 2;        // 2000
    (void)hid; (void)fin;

    float* ws      = (float*)d_ws;
    float* out_deg = ws;                         // [N]
    float* in_deg  = ws + N;                     // [N]
    float* h       = ws + 2 * (size_t)N;         // [N,64]
    float* agg     = h + (size_t)N * HID;        // [N,64]
    float* adj     = (float*)d_out;              // [N,N]
    float* rst     = adj + (size_t)N * N;        // [N,64] (second output)

    hipMemsetAsync(out_deg, 0, sizeof(float) * 2 * (size_t)N, stream);
    hipMemsetAsync(agg,     0, sizeof(float) * (size_t)N * HID, stream);

    deg_kernel<<<(E + 255) / 256, 256, 0, stream>>>(src, dst, out_deg, in_deg, E);

    // (N/16)*(HID/16) tiles, 8 waves per block -> exact grid, EXEC all ones
    proj_kernel<<<(N / 16) * (HID / 16) / 8, 256, 0, stream>>>(features, W, out_deg, h);

    edge_kernel<<<((size_t)E * 16 + 255) / 256, 256, 0, stream>>>(h, alpha, src, dst,
                                                                  node_id, agg, E, gene_num);

    norm_kernel<<<(N * HID) / 256, 256, 0, stream>>>(agg, in_deg, bias, rst);

    dim3 grid(N / 64, N / 128);                  // (256, 128)
    gemm_kernel<<<grid, 256, 0, stream>>>(rst, adj, N);
}